// ConvBlock_360777252979
// MI455X (gfx1250) — compile-verified
//
#include <hip/hip_runtime.h>

typedef __attribute__((ext_vector_type(2))) float v2f;
typedef __attribute__((ext_vector_type(8))) float v8f;

#define IN_CH   64
#define OUT_CH  128
#define HW      56
#define IMG_PIX (HW * HW)          // 3136
#define NIMG    32
#define KGEMM   (IN_CH * 9)        // 576
#define KSTEPS  (KGEMM / 4)        // 144
#define SLAB_W  58                 // 56 + halo
#define SLAB_ROWS (IN_CH * 3)      // 192 (ic, kh-row) rows
#define SLAB_ELEMS (SLAB_ROWS * SLAB_W)  // 11136 floats (~43.5 KB)
#define NPIX_TOT (NIMG * IMG_PIX)  // 100352 per channel

// d_ws float layout
#define WS_WFRAG   0        // 73728 floats: WMMA-A-fragment-ordered weights
#define WS_SUMS    73728    // 128 sums + 128 sum-of-squares
#define WS_PARAMS  73984    // 128 scale + 128 shift

// ---------------------------------------------------------------------------
// Compile-time k -> LDS-slab-offset table: off[k] = ic*3*58 + kh*58 + kw
// (k = ic*9 + kh*3 + kw). Read wave-uniformly as ushort4 per k-step.
// ---------------------------------------------------------------------------
struct alignas(16) KOffTab { unsigned short v[KGEMM]; };
constexpr KOffTab make_kofftab() {
    KOffTab t{};
    for (int k = 0; k < KGEMM; ++k) {
        int ic = k / 9, r = k % 9;
        t.v[k] = (unsigned short)(ic * (3 * SLAB_W) + (r / 3) * SLAB_W + (r % 3));
    }
    return t;
}
__constant__ KOffTab kofftab = make_kofftab();

static __device__ __forceinline__ v8f wmma_f32(v2f a, v2f b, v8f c) {
    return __builtin_amdgcn_wmma_f32_16x16x4_f32(false, a, false, b, (short)0, c,
                                                 false, false);
}

// ---------------------------------------------------------------------------
// Kernel 1: assemble block weights into WMMA A-fragment lane order.
// w_gemm[m][k] = w_blocks[m>>3][k>>3][m&7][k&7]   (P=16, Q=72, MB=8)
// frag idx = ((mtile*144 + kstep)*32 + lane); lane holds K = k0, k0+1 with
// k0 = kstep*4 + (lane>=16 ? 2 : 0), m = mtile*16 + (lane&15).
// Block 0 also zeroes the BN accumulators.
// ---------------------------------------------------------------------------
__global__ __launch_bounds__(256) void wprep_kernel(
    const float* __restrict__ wb, float* __restrict__ wfrag,
    float* __restrict__ sums)
{
    if (blockIdx.x == 0) sums[threadIdx.x] = 0.0f;  // 256 floats

    int idx = blockIdx.x * 256 + threadIdx.x;       // 8*144*32 = 36864 frags
    if (idx >= 8 * KSTEPS * 32) return;
    int lane  = idx & 31;
    int kstep = (idx >> 5) % KSTEPS;
    int mtile = idx / (KSTEPS * 32);
    int m  = mtile * 16 + (lane & 15);
    int k0 = kstep * 4 + ((lane >> 4) << 1);
    int k1 = k0 + 1;
    float2 v;
    v.x = wb[(((m >> 3) * 72 + (k0 >> 3)) * 8 + (m & 7)) * 8 + (k0 & 7)];
    v.y = wb[(((m >> 3) * 72 + (k1 >> 3)) * 8 + (m & 7)) * 8 + (k1 & 7)];
    reinterpret_cast<float2*>(wfrag)[idx] = v;
}

// ---------------------------------------------------------------------------
// Kernel 2: implicit-GEMM conv via V_WMMA_F32_16X16X4_F32.
// grid = 32 images * 56 rows; block = 256 (8 waves).
// Wave w -> M-tile pair (w>>1)*2..+1  x  N-tile pair (w&1)*2..+1,
// so each A and B fragment feeds 2 WMMAs. Fused BN sum/sum^2.
// Input slab staged via global_load_async_to_lds (ASYNCcnt path).
// ---------------------------------------------------------------------------
__global__ __launch_bounds__(256) void conv_wmma_kernel(
    const float* __restrict__ x, const float* __restrict__ wfrag,
    float* __restrict__ y, float* __restrict__ gsums)
{
    __shared__ float slab[SLAB_ELEMS];
    __shared__ float bn[256];

    const int tid = threadIdx.x;
    const int img = blockIdx.x / HW;
    const int row = blockIdx.x % HW;

    bn[tid] = 0.0f;

    // Stage input slab: 192 (ic, r) rows x 58 cols, zero-padded halo.
    // 8 lanes sweep one row; rowid strides by 32 -> 6 outer iterations.
    const float* xi = x + (size_t)img * (IN_CH * IMG_PIX);
    const int sub = tid & 7;
    for (int rowid = tid >> 3; rowid < SLAB_ROWS; rowid += 32) {
        int ic = rowid / 3;                    // only 6x per thread
        int r  = rowid - ic * 3;
        int gh = row + r - 1;
        bool rowok = (unsigned)gh < (unsigned)HW;
        const float* gp = xi + ic * IMG_PIX + gh * HW;
        float* lp = &slab[rowid * SLAB_W];
#pragma unroll
        for (int q = 0; q < 8; ++q) {
            int c = sub + 8 * q;               // 0..63
            if (c < SLAB_W) {
                int gw = c - 1;
                if (rowok && (unsigned)gw < (unsigned)HW) {
                    unsigned loff = (unsigned)(size_t)(lp + c);  // LDS byte off
                    const float* g = gp + gw;
                    asm volatile("global_load_async_to_lds_b32 %0, %1, off"
                                 :: "v"(loff), "v"(g) : "memory");
                } else {
                    lp[c] = 0.0f;
                }
            }
        }
    }
    asm volatile("s_wait_asynccnt 0" ::: "memory");
    __syncthreads();

    const int lane  = tid & 31;
    const int nlo   = lane & 15;
    const int khalf = lane >> 4;          // lanes 16-31 carry K+2,K+3
    const int wave  = tid >> 5;
    const int mp    = wave >> 1;          // M-tile pair 2mp, 2mp+1
    const int np    = wave & 1;           // N-tile pair 2np, 2np+1
    const int nbase = np * 32 + nlo;

    const v2f* frag = reinterpret_cast<const v2f*>(wfrag);
    const v2f* af0 = frag + ((2 * mp)     * KSTEPS) * 32 + lane;
    const v2f* af1 = frag + ((2 * mp + 1) * KSTEPS) * 32 + lane;
    const ushort4* otab = reinterpret_cast<const ushort4*>(kofftab.v);

    v8f acc00 = {}, acc01 = {}, acc10 = {}, acc11 = {};

    for (int kso = 0; kso < KSTEPS; kso += 4) {
        __builtin_prefetch(af0 + (kso + 16) * 32, 0, 0);  // global_prefetch_b8
        __builtin_prefetch(af1 + (kso + 16) * 32, 0, 0);
#pragma unroll
        for (int ku = 0; ku < 4; ++ku) {
            const int ks = kso + ku;
            v2f a0 = af0[ks * 32];
            v2f a1 = af1[ks * 32];

            ushort4 o = otab[ks];             // wave-uniform -> s_load
            int b0o = (khalf ? (int)o.z : (int)o.x) + nbase;
            int b1o = (khalf ? (int)o.w : (int)o.y) + nbase;

            v2f bA, bB;
            bA.x = slab[b0o];      bA.y = slab[b1o];
            bB.x = slab[b0o + 16]; bB.y = slab[b1o + 16];

            acc00 = wmma_f32(a0, bA, acc00);
            acc01 = wmma_f32(a0, bB, acc01);
            acc10 = wmma_f32(a1, bA, acc10);
            acc11 = wmma_f32(a1, bB, acc11);
        }
    }

    // Epilogue: store D tiles + fused BN partials.
    // C/D layout: VGPR j -> M = j (lanes 0-15) / j+8 (lanes 16-31), N = lane&15.
    float* yo = y + (size_t)img * (OUT_CH * IMG_PIX) + row * HW;
    v8f accm[2][2] = {{acc00, acc01}, {acc10, acc11}};

#pragma unroll
    for (int mi = 0; mi < 2; ++mi) {
        const int mbase = (2 * mp + mi) * 16 + khalf * 8;
#pragma unroll
        for (int j = 0; j < 8; ++j) {
            float s = 0.0f, s2 = 0.0f;
#pragma unroll
            for (int ni = 0; ni < 2; ++ni) {
                int wcol = (2 * np + ni) * 16 + nlo;
                float v = accm[mi][ni][j];
                if (wcol < HW) {
                    yo[(size_t)(mbase + j) * IMG_PIX + wcol] = v;
                    s  += v;
                    s2 += v * v;
                }
            }
            atomicAdd(&bn[mbase + j],       s);   // ds_add_f32
            atomicAdd(&bn[128 + mbase + j], s2);
        }
    }
    __syncthreads();
    atomicAdd(&gsums[tid], bn[tid]);   // one global f32 atomic per entry per WG
}

// ---------------------------------------------------------------------------
// Kernel 3: fold mean/var + gamma/beta into (scale, shift).
// ---------------------------------------------------------------------------
__global__ void bn_finalize_kernel(const float* __restrict__ sums,
                                   const float* __restrict__ gamma,
                                   const float* __restrict__ beta,
                                   float* __restrict__ params)
{
    int c = threadIdx.x;                      // 128 threads
    const float n = (float)NPIX_TOT;
    float mean = sums[c] / n;
    float var  = sums[128 + c] / n - mean * mean;
    float inv  = gamma[c] * rsqrtf(var + 1e-5f);
    params[c]       = inv;
    params[128 + c] = beta[c] - mean * inv;
}

// ---------------------------------------------------------------------------
// Kernel 4: in-place affine + clip(0,6), float4-vectorized.
// ---------------------------------------------------------------------------
__global__ __launch_bounds__(256) void bn_apply_kernel(
    float* __restrict__ y, const float* __restrict__ params)
{
    const int total4 = NIMG * OUT_CH * IMG_PIX / 4;     // 3,211,264
    int i = blockIdx.x * 256 + threadIdx.x;
    if (i >= total4) return;
    int ch = (i / (IMG_PIX / 4)) & (OUT_CH - 1);
    float inv = params[ch];
    float sh  = params[128 + ch];
    float4 v = reinterpret_cast<float4*>(y)[i];
    v.x = fminf(fmaxf(v.x * inv + sh, 0.0f), 6.0f);
    v.y = fminf(fmaxf(v.y * inv + sh, 0.0f), 6.0f);
    v.z = fminf(fmaxf(v.z * inv + sh, 0.0f), 6.0f);
    v.w = fminf(fmaxf(v.w * inv + sh, 0.0f), 6.0f);
    reinterpret_cast<float4*>(y)[i] = v;
}

// ---------------------------------------------------------------------------
extern "C" void kernel_launch(void* const* d_in, const int* in_sizes, int n_in,
                              void* d_out, int out_size, void* d_ws, size_t ws_size,
                              hipStream_t stream)
{
    const float* x     = (const float*)d_in[0];
    const float* wb    = (const float*)d_in[1];
    const float* gamma = (const float*)d_in[2];
    const float* beta  = (const float*)d_in[3];
    float* y  = (float*)d_out;
    float* ws = (float*)d_ws;

    float* wfrag  = ws + WS_WFRAG;
    float* sums   = ws + WS_SUMS;
    float* params = ws + WS_PARAMS;

    wprep_kernel<<<(8 * KSTEPS * 32 + 255) / 256, 256, 0, stream>>>(wb, wfrag, sums);
    conv_wmma_kernel<<<NIMG * HW, 256, 0, stream>>>(x, wfrag, y, sums);
    bn_finalize_kernel<<<1, 128, 0, stream>>>(sums, gamma, beta, params);

    const int total4 = NIMG * OUT_CH * IMG_PIX / 4;
    bn_apply_kernel<<<(total4 + 255) / 256, 256, 0, stream>>>(y, params);
}